// CIPL_8306466750573
// MI455X (gfx1250) — compile-verified
//
#include <hip/hip_runtime.h>

// ---------------------------------------------------------------------------
// Problem constants (from the reference)
// ---------------------------------------------------------------------------
#define N8     8
#define N16    16
#define CCH    128     // channels (K for distance GEMMs)
#define HWSZ   784     // 28*28, divisible by 16 (49 tiles)
#define PP     2000    // prototypes, divisible by 16 (125 tiles)
#define NCLS   200

typedef float v2f __attribute__((ext_vector_type(2)));
typedef float v8f __attribute__((ext_vector_type(8)));

// One V_WMMA_F32_16X16X4_F32: D(16x16,f32) = A(16x4,f32) * B(4x16,f32) + C
__device__ __forceinline__ v8f wmma_k4(v2f a, v2f b, v8f c) {
    return __builtin_amdgcn_wmma_f32_16x16x4_f32(false, a, false, b,
                                                 (short)0, c, false, false);
}

// ---------------------------------------------------------------------------
// Small helper kernels
// ---------------------------------------------------------------------------
__global__ void k_fill(float* __restrict__ p, int n, float v) {
    int i = blockIdx.x * blockDim.x + threadIdx.x;
    if (i < n) p[i] = v;
}

// p2[p] = sum_c proto[p,c]^2     (proto is (P,C) row-major)
__global__ void k_p2(const float* __restrict__ proto, float* __restrict__ p2) {
    int p = blockIdx.x * blockDim.x + threadIdx.x;
    if (p >= PP) return;
    const float* r = proto + (size_t)p * CCH;
    float s = 0.f;
    #pragma unroll 8
    for (int c = 0; c < CCH; ++c) s += r[c] * r[c];
    p2[p] = s;
}

// x2[n,hw] = sum_c x[n,c,hw]^2   (x is (nb,C,HW)); coalesced across hw
__global__ void k_x2(const float* __restrict__ x, float* __restrict__ x2, int nb) {
    int idx = blockIdx.x * blockDim.x + threadIdx.x;
    if (idx >= nb * HWSZ) return;
    int n = idx / HWSZ, hw = idx % HWSZ;
    const float* xb = x + ((size_t)n * CCH) * HWSZ + hw;
    float s = 0.f;
    for (int c = 0; c < CCH; ++c) { float v = xb[(size_t)c * HWSZ]; s += v * v; }
    x2[idx] = s;
}

// ---------------------------------------------------------------------------
// L2-distance classifier core (WMMA):
//   d[n,p,hw] = relu(x2[n,hw] - 2*sum_c proto[p,c]*x[n,c,hw] + p2[p])
//   sim       = exp(-d/256)            (compile-time optional, NT stores)
//   md[n,p]   = min_hw d               (wave shuffle + bitwise atomicMin, d>=0)
// One wave computes a 16(P) x 16(HW) tile; 8 waves / block; blockIdx.y = n.
// ---------------------------------------------------------------------------
template <bool DO_SIM>
__global__ __launch_bounds__(256)
void k_l2(const float* __restrict__ x,      // (nb,C,HW)
          const float* __restrict__ x2,     // (nb,HW)
          const float* __restrict__ proto,  // (P,C)
          const float* __restrict__ p2,     // (P)
          float* __restrict__ sim,          // (nb,P,HW) (unused if !DO_SIM)
          float* __restrict__ md)           // (nb,P), pre-filled +BIG
{
    const int n    = blockIdx.y;
    const int wave = threadIdx.x >> 5;
    const int lane = threadIdx.x & 31;
    const int PT = PP / 16, HT = HWSZ / 16;           // 125, 49
    const int tile = blockIdx.x * 8 + wave;
    if (tile >= PT * HT) return;                       // wave-uniform guard
    const int p0   = (tile / HT) * 16;
    const int h0   = (tile % HT) * 16;
    const int half = lane >> 4;                        // K-half select
    const int l16  = lane & 15;

    const float* xb   = x + (size_t)n * CCH * HWSZ;
    const float* arow = proto + (size_t)(p0 + l16) * CCH + 2 * half; // A: M=p0+l16
    const float* bcol = xb + (size_t)(2 * half) * HWSZ + (h0 + l16); // B: N=h0+l16

    v8f acc = {0.f,0.f,0.f,0.f,0.f,0.f,0.f,0.f};
    #pragma unroll 4
    for (int k = 0; k < CCH; k += 4) {
        v2f a = *(const v2f*)(arow + k);               // K = k+2h, k+2h+1 (8B aligned)
        v2f b;
        b.x = bcol[(size_t)k * HWSZ];
        b.y = bcol[(size_t)(k + 1) * HWSZ];
        acc = wmma_k4(a, b, acc);
    }

    // Epilogue: lane element v -> (M = p0 + v + 8*half, N = h0 + l16).
    // Row-base pointers so the 8 stores/loads use immediate offsets (v*3136B).
    const int   prow  = p0 + 8 * half;                 // first P row of this half
    const float x2c   = x2[(size_t)n * HWSZ + h0 + l16];
    const float* p2r  = p2 + prow;
    float* simr = DO_SIM ? (sim + ((size_t)n * PP + prow) * HWSZ + (h0 + l16))
                         : nullptr;
    float dv[8];
    #pragma unroll
    for (int v = 0; v < 8; ++v) {
        float d = x2c - 2.0f * acc[v] + p2r[v];
        d = d > 0.f ? d : 0.f;
        dv[v] = d;
        if (DO_SIM) {
            __builtin_nontemporal_store(__expf(d * (-1.0f / 256.0f)),
                                        simr + (size_t)v * HWSZ);
        }
    }
    // min over the 16 columns held by each lane-half (lanes share a row set)
    float* mdr = md + (size_t)n * PP + prow;
    #pragma unroll
    for (int v = 0; v < 8; ++v) {
        float m = dv[v];
        #pragma unroll
        for (int off = 1; off < 16; off <<= 1)
            m = fminf(m, __shfl_xor(m, off, 32));
        if (l16 == 0)
            atomicMin((unsigned int*)(mdr + v), __float_as_uint(m)); // valid: d >= 0
    }
}

// logits[n,cls] = sum_p exp(-md[n,p]/256) * w[cls,p]   (tiny: 16x200x2000)
__global__ void k_logits(const float* __restrict__ md, const float* __restrict__ w,
                         float* __restrict__ out, int rows) {
    int idx = blockIdx.x * blockDim.x + threadIdx.x;
    if (idx >= rows * NCLS) return;
    int n = idx / NCLS, cls = idx % NCLS;
    const float* mdr = md + (size_t)n * PP;
    const float* wr  = w  + (size_t)cls * PP;
    float s = 0.f;
    for (int p = 0; p < PP; ++p)
        s += __expf(mdr[p] * (-1.0f / 256.0f)) * wr[p];
    out[idx] = s;
}

// ---------------------------------------------------------------------------
// A2[n,i,j] = sum_c f1[n,c,i] * f2[n,c,j]   (784x784x128 per n, WMMA)
// ---------------------------------------------------------------------------
__global__ __launch_bounds__(256)
void k_a2(const float* __restrict__ f1, const float* __restrict__ f2,
          float* __restrict__ A2) {
    const int n    = blockIdx.y;
    const int wave = threadIdx.x >> 5;
    const int lane = threadIdx.x & 31;
    const int T = HWSZ / 16;                           // 49
    const int tile = blockIdx.x * 8 + wave;
    if (tile >= T * T) return;
    const int i0 = (tile / T) * 16, j0 = (tile % T) * 16;
    const int half = lane >> 4, l16 = lane & 15;

    const float* f1b = f1 + (size_t)n * CCH * HWSZ + (size_t)(2 * half) * HWSZ + (i0 + l16);
    const float* f2b = f2 + (size_t)n * CCH * HWSZ + (size_t)(2 * half) * HWSZ + (j0 + l16);

    v8f acc = {0.f,0.f,0.f,0.f,0.f,0.f,0.f,0.f};
    #pragma unroll 4
    for (int k = 0; k < CCH; k += 4) {
        v2f a, b;
        a.x = f1b[(size_t)k * HWSZ];       a.y = f1b[(size_t)(k + 1) * HWSZ];
        b.x = f2b[(size_t)k * HWSZ];       b.y = f2b[(size_t)(k + 1) * HWSZ];
        acc = wmma_k4(a, b, acc);
    }
    float* o = A2 + ((size_t)n * HWSZ + (i0 + 8 * half)) * HWSZ + (j0 + l16);
    #pragma unroll
    for (int v = 0; v < 8; ++v)
        o[(size_t)v * HWSZ] = acc[v];
}

// Row stats (softmax over j): one wave per row (n,i); contiguous 32-wide reads
__global__ __launch_bounds__(256)
void k_rowstats(const float* __restrict__ A2, float* __restrict__ rmax,
                float* __restrict__ rsum) {
    const int wave = threadIdx.x >> 5, lane = threadIdx.x & 31;
    const int row = blockIdx.x * 8 + wave;
    if (row >= N8 * HWSZ) return;
    const float* r = A2 + (size_t)row * HWSZ;
    float m = -3.0e38f;
    for (int j = lane; j < HWSZ; j += 32) m = fmaxf(m, r[j]);
    #pragma unroll
    for (int off = 16; off; off >>= 1) m = fmaxf(m, __shfl_xor(m, off, 32));
    float s = 0.f;
    for (int j = lane; j < HWSZ; j += 32) s += __expf(r[j] - m);
    #pragma unroll
    for (int off = 16; off; off >>= 1) s += __shfl_xor(s, off, 32);
    if (lane == 0) { rmax[row] = m; rsum[row] = s; }
}

// Col stats (softmax over i): one thread per (n,j); reads coalesced across j
__global__ void k_colstats(const float* __restrict__ A2, float* __restrict__ cmax,
                           float* __restrict__ csum) {
    int idx = blockIdx.x * blockDim.x + threadIdx.x;
    if (idx >= N8 * HWSZ) return;
    const int n = idx / HWSZ, j = idx % HWSZ;
    const float* col = A2 + (size_t)n * HWSZ * HWSZ + j;
    float m = -3.0e38f;
    for (int i = 0; i < HWSZ; ++i) m = fmaxf(m, col[(size_t)i * HWSZ]);
    float s = 0.f;
    for (int i = 0; i < HWSZ; ++i) s += __expf(col[(size_t)i * HWSZ] - m);
    cmax[idx] = m; csum[idx] = s;
}

// Bmat[n,j,i] = exp(A2[n,i,j]-rmax[n,i]) / rsum[n,i]   (transposed softmax)
__global__ void k_makeB(const float* __restrict__ A2, const float* __restrict__ rmax,
                        const float* __restrict__ rsum, float* __restrict__ B) {
    int idx = blockIdx.x * blockDim.x + threadIdx.x;
    if (idx >= N8 * HWSZ * HWSZ) return;
    const int n = idx / (HWSZ * HWSZ), rem = idx % (HWSZ * HWSZ);
    const int j = rem / HWSZ, i = rem % HWSZ;
    const float v = A2[((size_t)n * HWSZ + i) * HWSZ + j];
    B[idx] = __expf(v - rmax[n * HWSZ + i]) / rsum[n * HWSZ + i];
}

// A (in-place over A2): A[n,i,j] = exp(A2-cmax[n,j]) / csum[n,j]
__global__ void k_makeA(float* __restrict__ A2, const float* __restrict__ cmax,
                        const float* __restrict__ csum) {
    int idx = blockIdx.x * blockDim.x + threadIdx.x;
    if (idx >= N8 * HWSZ * HWSZ) return;
    const int n = idx / (HWSZ * HWSZ), j = idx % HWSZ;
    A2[idx] = __expf(A2[idx] - cmax[n * HWSZ + j]) / csum[n * HWSZ + j];
}

// out[n,c,j] = sum_i X[n,c,i] * M[n,i,j]   (128 x 784 x 784 per n, WMMA)
__global__ __launch_bounds__(256)
void k_att(const float* __restrict__ X, const float* __restrict__ Mx,
           float* __restrict__ out) {
    const int n    = blockIdx.y;
    const int wave = threadIdx.x >> 5;
    const int lane = threadIdx.x & 31;
    const int CT = CCH / 16, JT = HWSZ / 16;           // 8, 49
    const int tile = blockIdx.x * 8 + wave;
    if (tile >= CT * JT) return;
    const int c0 = (tile / JT) * 16, j0 = (tile % JT) * 16;
    const int half = lane >> 4, l16 = lane & 15;

    const float* Xr  = X  + ((size_t)n * CCH + (c0 + l16)) * HWSZ + 2 * half;
    const float* Mxb = Mx + (size_t)n * HWSZ * HWSZ + (size_t)(2 * half) * HWSZ + (j0 + l16);

    v8f acc = {0.f,0.f,0.f,0.f,0.f,0.f,0.f,0.f};
    #pragma unroll 2
    for (int k = 0; k < HWSZ; k += 4) {
        v2f a = *(const v2f*)(Xr + k);                 // contiguous K pair
        v2f b;
        b.x = Mxb[(size_t)k * HWSZ];
        b.y = Mxb[(size_t)(k + 1) * HWSZ];
        acc = wmma_k4(a, b, acc);
    }
    float* o = out + ((size_t)n * CCH + (c0 + 8 * half)) * HWSZ + (j0 + l16);
    #pragma unroll
    for (int v = 0; v < 8; ++v)
        o[(size_t)v * HWSZ] = acc[v];
}

// ---------------------------------------------------------------------------
// Host-side orchestration
// ---------------------------------------------------------------------------
extern "C" void kernel_launch(void* const* d_in, const int* in_sizes, int n_in,
                              void* d_out, int out_size, void* d_ws, size_t ws_size,
                              hipStream_t stream) {
    const float* f1    = (const float*)d_in[0];
    const float* f2    = (const float*)d_in[1];
    const float* faug  = (const float*)d_in[2];
    const float* proto = (const float*)d_in[3];
    const float* protoE= (const float*)d_in[4];
    const float* wlast = (const float*)d_in[5];
    const float* wlastE= (const float*)d_in[6];
    float* out = (float*)d_out;

    // output offsets (floats)
    const size_t o_score = 0;
    const size_t o_md    = o_score + (size_t)N16 * NCLS;       // 3200
    const size_t o_co    = o_md    + (size_t)N16 * PP;         // 35200
    const size_t o_sim   = o_co    + (size_t)N16 * NCLS;       // 38400
    const size_t o_sime  = o_sim   + (size_t)N16 * PP * HWSZ;  // +25,088,000
    const size_t o_loge  = o_sime  + (size_t)N16 * PP * HWSZ;

    // workspace layout (floats)
    float* ws = (float*)d_ws;
    float* w_p2    = ws;                 ws += PP;
    float* w_p2e   = ws;                 ws += PP;
    float* w_x2f1  = ws;                 ws += N8  * HWSZ;
    float* w_x2f2  = ws;                 ws += N8  * HWSZ;
    float* w_x2aug = ws;                 ws += N16 * HWSZ;
    float* w_x2a1  = ws;                 ws += N8  * HWSZ;
    float* w_x2a2  = ws;                 ws += N8  * HWSZ;
    float* w_mdatt = ws;                 ws += (size_t)N16 * PP;   // a1 rows 0-7, a2 rows 8-15
    float* w_mdaug = ws;                 ws += (size_t)N16 * PP;
    float* w_A2    = ws;                 ws += (size_t)N8 * HWSZ * HWSZ; // becomes Amat in-place
    float* w_B     = ws;                 ws += (size_t)N8 * HWSZ * HWSZ;
    float* w_f1att = ws;                 ws += (size_t)N8 * CCH * HWSZ;
    float* w_f2att = ws;                 ws += (size_t)N8 * CCH * HWSZ;
    float* w_rmax  = ws;                 ws += N8 * HWSZ;
    float* w_rsum  = ws;                 ws += N8 * HWSZ;
    float* w_cmax  = ws;                 ws += N8 * HWSZ;
    float* w_csum  = ws;                 ws += N8 * HWSZ;

    const int TPB = 256;
    const int l2_blocks = (125 * 49 + 7) / 8;          // 766
    const int a2_blocks = (49 * 49 + 7) / 8;           // 301
    const int at_blocks = (8 * 49 + 7) / 8;            // 49
    const int nelemA2   = N8 * HWSZ * HWSZ;

    // 1) norms
    k_p2<<<(PP + TPB - 1) / TPB, TPB, 0, stream>>>(proto,  w_p2);
    k_p2<<<(PP + TPB - 1) / TPB, TPB, 0, stream>>>(protoE, w_p2e);
    k_x2<<<(N8  * HWSZ + TPB - 1) / TPB, TPB, 0, stream>>>(f1,   w_x2f1,  N8);
    k_x2<<<(N8  * HWSZ + TPB - 1) / TPB, TPB, 0, stream>>>(f2,   w_x2f2,  N8);
    k_x2<<<(N16 * HWSZ + TPB - 1) / TPB, TPB, 0, stream>>>(faug, w_x2aug, N16);

    // 2) init min-distance accumulators
    k_fill<<<(N16 * PP + TPB - 1) / TPB, TPB, 0, stream>>>(out + o_md, N16 * PP, 3.0e38f);
    k_fill<<<(N16 * PP + TPB - 1) / TPB, TPB, 0, stream>>>(w_mdatt,    N16 * PP, 3.0e38f);
    k_fill<<<(N16 * PP + TPB - 1) / TPB, TPB, 0, stream>>>(w_mdaug,    N16 * PP, 3.0e38f);

    // 3) main classifiers (sim1/sim2 -> out, simi_ema -> out)
    k_l2<true><<<dim3(l2_blocks, N8),  TPB, 0, stream>>>(f1,   w_x2f1,  proto,  w_p2,
                                                         out + o_sim, out + o_md);
    k_l2<true><<<dim3(l2_blocks, N8),  TPB, 0, stream>>>(f2,   w_x2f2,  proto,  w_p2,
                                                         out + o_sim + (size_t)N8 * PP * HWSZ,
                                                         out + o_md + (size_t)N8 * PP);
    k_l2<true><<<dim3(l2_blocks, N16), TPB, 0, stream>>>(faug, w_x2aug, protoE, w_p2e,
                                                         out + o_sime, w_mdaug);

    // 4) logits from min distances
    k_logits<<<(N16 * NCLS + TPB - 1) / TPB, TPB, 0, stream>>>(out + o_md, wlast,
                                                               out + o_score, N16);
    k_logits<<<(N16 * NCLS + TPB - 1) / TPB, TPB, 0, stream>>>(w_mdaug, wlastE,
                                                               out + o_loge, N16);

    // 5) co-attention
    k_a2<<<dim3(a2_blocks, N8), TPB, 0, stream>>>(f1, f2, w_A2);
    k_rowstats<<<HWSZ, TPB, 0, stream>>>(w_A2, w_rmax, w_rsum);            // 8*784/8 waves
    k_colstats<<<(N8 * HWSZ + TPB - 1) / TPB, TPB, 0, stream>>>(w_A2, w_cmax, w_csum);
    k_makeB<<<(nelemA2 + TPB - 1) / TPB, TPB, 0, stream>>>(w_A2, w_rmax, w_rsum, w_B);
    k_makeA<<<(nelemA2 + TPB - 1) / TPB, TPB, 0, stream>>>(w_A2, w_cmax, w_csum); // in-place
    k_att<<<dim3(at_blocks, N8), TPB, 0, stream>>>(f1, w_A2, w_f2att);     // f2_att = f1 @ A
    k_att<<<dim3(at_blocks, N8), TPB, 0, stream>>>(f2, w_B,  w_f1att);     // f1_att = f2 @ B

    // 6) classifiers on attended features (logits only)
    k_x2<<<(N8 * HWSZ + TPB - 1) / TPB, TPB, 0, stream>>>(w_f1att, w_x2a1, N8);
    k_x2<<<(N8 * HWSZ + TPB - 1) / TPB, TPB, 0, stream>>>(w_f2att, w_x2a2, N8);
    k_l2<false><<<dim3(l2_blocks, N8), TPB, 0, stream>>>(w_f1att, w_x2a1, proto, w_p2,
                                                         nullptr, w_mdatt);
    k_l2<false><<<dim3(l2_blocks, N8), TPB, 0, stream>>>(w_f2att, w_x2a2, proto, w_p2,
                                                         nullptr, w_mdatt + (size_t)N8 * PP);
    k_logits<<<(N16 * NCLS + TPB - 1) / TPB, TPB, 0, stream>>>(w_mdatt, wlast,
                                                               out + o_co, N16);
    (void)in_sizes; (void)n_in; (void)out_size; (void)ws_size;
}